// VarPointMVSNetReg_82721070120980
// MI455X (gfx1250) — compile-verified
//
#include <hip/hip_runtime.h>

// ---------------------------------------------------------------------------
// VarPointMVSNet for MI455X (gfx1250, wave32).
// fp32 WMMA (v_wmma_f32_16x16x4_f32) implicit-GEMM for the conv2d tower with
// compile-time-unrolled K windows, LDS-staged zero-padded weight tiles,
// dual N-tiles per wave, and software-pipelined (double-buffered) A loads.
// VALU kernels for warp/variance/3D-UNet/softmax.
// ---------------------------------------------------------------------------

typedef float v2f __attribute__((ext_vector_type(2)));
typedef float v8f __attribute__((ext_vector_type(8)));

#define DNUM   48
#define FHH    64
#define FWW    80
#define NPIXF  (FHH * FWW)     // 5120
#define NVIEW  3
#define NPTS   (DNUM * NPIXF)  // 245760

static __device__ __forceinline__ int imin(int a, int b) { return a < b ? a : b; }
static __device__ __forceinline__ int imax(int a, int b) { return a > b ? a : b; }

// ---------------------------------------------------------------------------
// Camera setup: invert K0 / R0, stash per-view R|t|K (halved fx,fy rows).
// CAM layout (floats): [0..8] Rinv0, [9..17] Kinv0, [18] dstart, [19] dint,
//   then per view v: base=20+21v : R(9), t(3), K(9)
// ---------------------------------------------------------------------------
static __device__ void inv3x3(const float* m, float* o) {
    float a = m[0], b = m[1], c = m[2];
    float d = m[3], e = m[4], f = m[5];
    float g = m[6], h = m[7], i = m[8];
    float A =  (e * i - f * h);
    float Bc = -(d * i - f * g);
    float Cc =  (d * h - e * g);
    float det = a * A + b * Bc + c * Cc;
    float r = 1.0f / det;
    o[0] = A * r;               o[1] = -(b * i - c * h) * r; o[2] =  (b * f - c * e) * r;
    o[3] = Bc * r;              o[4] =  (a * i - c * g) * r; o[5] = -(a * f - c * d) * r;
    o[6] = Cc * r;              o[7] = -(a * h - b * g) * r; o[8] =  (a * e - b * d) * r;
}

__global__ void camera_setup_kernel(const float* __restrict__ cam, float* __restrict__ C) {
    if (threadIdx.x != 0 || blockIdx.x != 0) return;
    float R0[9], K0[9];
    for (int v = 0; v < NVIEW; ++v) {
        const float* E  = cam + v * 32;        // extrinsic 4x4
        const float* Kc = cam + v * 32 + 16;   // intrinsic 4x4
        float* base = C + 20 + v * 21;
        for (int r = 0; r < 3; ++r)
            for (int c = 0; c < 3; ++c)
                base[r * 3 + c] = E[r * 4 + c];
        for (int r = 0; r < 3; ++r)
            base[9 + r] = E[r * 4 + 3];
        for (int r = 0; r < 3; ++r)
            for (int c = 0; c < 3; ++c)
                base[12 + r * 3 + c] = Kc[r * 4 + c] * (r < 2 ? 0.5f : 1.0f);
        if (v == 0) {
            for (int j = 0; j < 9; ++j) { R0[j] = base[j]; K0[j] = base[12 + j]; }
        }
    }
    inv3x3(R0, C + 0);
    inv3x3(K0, C + 9);
    C[18] = cam[16 + 12];  // view0, mat1, [3][0] : depth_start
    C[19] = cam[16 + 13];  // view0, mat1, [3][1] : depth_interval
}

// ---------------------------------------------------------------------------
// World points (output tensor #0): X = Rinv0 * (Kinv0*[x+.5,y+.5,1]*depth - t0)
// ---------------------------------------------------------------------------
__global__ void world_points_kernel(const float* __restrict__ C, float* __restrict__ wp) {
    int n = blockIdx.x * blockDim.x + threadIdx.x;
    if (n >= NPTS) return;
    int d = n / NPIXF;
    int pix = n - d * NPIXF;
    int y = pix / FWW;
    int x = pix - y * FWW;
    const float* Rinv = C;
    const float* Kinv = C + 9;
    float dstart = C[18], dint = C[19];
    float px = (float)x + 0.5f, py = (float)y + 0.5f;
    float ux = Kinv[0] * px + Kinv[1] * py + Kinv[2];
    float uy = Kinv[3] * px + Kinv[4] * py + Kinv[5];
    float uz = Kinv[6] * px + Kinv[7] * py + Kinv[8];
    float depth = dstart + (float)d * dint;
    const float* t0 = C + 20 + 9;
    float cx = ux * depth - t0[0];
    float cy = uy * depth - t0[1];
    float cz = uz * depth - t0[2];
    wp[0 * NPTS + n] = Rinv[0] * cx + Rinv[1] * cy + Rinv[2] * cz;
    wp[1 * NPTS + n] = Rinv[3] * cx + Rinv[4] * cy + Rinv[5] * cz;
    wp[2 * NPTS + n] = Rinv[6] * cx + Rinv[7] * cy + Rinv[8] * cz;
}

// ---------------------------------------------------------------------------
// Direct conv2d for narrow layers (3->8, 8->8): one thread per output element.
// ---------------------------------------------------------------------------
__global__ void conv2d_direct_kernel(const float* __restrict__ in, const float* __restrict__ w,
                                     float* __restrict__ out, int nimg, int IC, int OC,
                                     int IH, int IW, int K, int relu) {
    long idx = (long)blockIdx.x * blockDim.x + threadIdx.x;
    long per = (long)OC * IH * IW;
    long tot = (long)nimg * per;
    if (idx >= tot) return;
    int img = (int)(idx / per);
    long r0 = idx - (long)img * per;
    int oc = (int)(r0 / (IH * IW));
    int p  = (int)(r0 - (long)oc * IH * IW);
    int oy = p / IW, ox = p - oy * IW;
    int pad = K / 2;
    const float* inb = in + (long)img * IC * IH * IW;
    float acc = 0.0f;
    for (int ic = 0; ic < IC; ++ic) {
        const float* wb = w + ((long)oc * IC + ic) * K * K;
        const float* ib = inb + (long)ic * IH * IW;
        for (int ky = 0; ky < K; ++ky) {
            int yi = oy + ky - pad;
            if (yi < 0 || yi >= IH) continue;
            for (int kx = 0; kx < K; ++kx) {
                int xi = ox + kx - pad;
                if (xi < 0 || xi >= IW) continue;
                acc = fmaf(ib[(long)yi * IW + xi], wb[ky * K + kx], acc);
            }
        }
    }
    if (relu) acc = fmaxf(acc, 0.0f);
    out[(long)img * per + r0] = acc;
}

// ---------------------------------------------------------------------------
// Implicit-GEMM conv2d with V_WMMA_F32_16X16X4_F32 (fp32 in, fp32 acc).
//   M = 16 output pixels / wave (8 waves per block), N = NT*16 oc / block,
//   K = IC * KKP where KKP = K*K rounded up to a multiple of 4 (weights zero-
//   padded in LDS so the pad taps contribute nothing).
// Per-lane tap offsets/masks depend only on the step index, NOT on ic, so the
// inner ic-loop carries zero address arithmetic. A-taps for channel ic+1 are
// batch-loaded (32-bit offsets, SADDR-friendly) while channel ic's WMMAs
// consume the previous batch: one full channel of matrix work hides the
// global-load latency. Wave-uniform early exit keeps EXEC all-ones for WMMA.
// Fragment layouts per CDNA5 ISA 7.12.2:
//   A (16x4): lane -> (M=lane&15), K-half = lane>>4; a.x/a.y = the half's rows
//   B (4x16): lane -> (N=lane&15), K-half = lane>>4
//   C/D (16x16): VGPR r -> row r + 8*(lane>>4), col = lane&15
// ---------------------------------------------------------------------------
template <int K, int NT>
__global__ void __launch_bounds__(256)
conv2d_wmma_kernel(const float* __restrict__ in, const float* __restrict__ w,
                   float* __restrict__ out, int IC, int IH, int IW,
                   int OH, int OW, int stride, int relu) {
    constexpr int KK    = K * K;
    constexpr int KKP   = (KK + 3) & ~3;   // 12 (3x3) or 28 (5x5)
    constexpr int NSTEP = KKP / 4;         // 3 or 7
    constexpr int NC    = 16 * NT;         // out-channels per block
    constexpr int PAD   = K / 2;
    extern __shared__ float ldsw[];        // [IC*KKP][NC], zero-padded taps

    const int OC   = gridDim.y * NC;
    const int ocb  = blockIdx.y * NC;
    const int img  = blockIdx.z;
    const int npix = OH * OW;
    const int chan = IH * IW;
    const float* inb = in + (long)img * IC * chan;
    float* outb = out + (long)img * OC * npix;

    // Cooperative stage of the weight tile into LDS: ldsw[(ic*KKP + r)*NC + n]
    // w is (OC, IC, K, K) contiguous; pad rows (r >= KK) are zero.
    for (int idx = threadIdx.x; idx < IC * KKP * NC; idx += blockDim.x) {
        int n  = idx & (NC - 1);
        int kr = idx / NC;
        int ic = kr / KKP;
        int r  = kr - ic * KKP;
        ldsw[idx] = (r < KK) ? w[((long)(ocb + n) * IC + ic) * KK + r] : 0.0f;
    }
    __syncthreads();

    const int lane  = threadIdx.x & 31;
    const int wave  = threadIdx.x >> 5;
    const int mbase = (blockIdx.x * 8 + wave) * 16;
    if (mbase >= npix) return;             // wave-uniform: EXEC stays all-ones

    const int mcol = lane & 15;            // A row / D col within tile
    const int half = lane >> 4;            // K-half select
    const int p    = mbase + mcol;
    const bool pv  = p < npix;
    const int oy   = p / OW;
    const int ox   = p - oy * OW;

    // Precompute per-step tap offsets (clamped, always in-bounds) and masks.
    // 32-bit element offsets -> SADDR + voffset global loads.
    int  off[NSTEP][2];
    bool okm[NSTEP][2];
#pragma unroll
    for (int s = 0; s < NSTEP; ++s) {
#pragma unroll
        for (int j = 0; j < 2; ++j) {
            int r  = s * 4 + half * 2 + j;     // position within KKP window
            int ky = (r < KK) ? (r / K) : 0;   // const-divisor -> mul-magic
            int kx = (r < KK) ? (r - ky * K) : 0;
            int yi = oy * stride + ky - PAD;
            int xi = ox * stride + kx - PAD;
            okm[s][j] = pv && (r < KK) && yi >= 0 && yi < IH && xi >= 0 && xi < IW;
            off[s][j] = imin(imax(yi, 0), IH - 1) * IW + imin(imax(xi, 0), IW - 1);
        }
    }

    // Warm caches with this lane's first input row (global_prefetch_b8).
    __builtin_prefetch(inb + off[0][0], 0, 1);

    v8f acc[NT];
#pragma unroll
    for (int t = 0; t < NT; ++t) acc[t] = v8f{0.f, 0.f, 0.f, 0.f, 0.f, 0.f, 0.f, 0.f};

    // Software pipeline: batch-load all taps for channel ic+1 while the WMMAs
    // consume channel ic's registers (last iter clamps to a harmless reload).
    float va[NSTEP][2];
#pragma unroll
    for (int s = 0; s < NSTEP; ++s) {
        va[s][0] = inb[off[s][0]];
        va[s][1] = inb[off[s][1]];
    }

    int icoff = 0;
    int lbase = 0;
    for (int ic = 0; ic < IC; ++ic) {
        const int icoff_n = (ic + 1 < IC) ? icoff + chan : icoff;
        float vb[NSTEP][2];
#pragma unroll
        for (int s = 0; s < NSTEP; ++s) {
            vb[s][0] = inb[icoff_n + off[s][0]];
            vb[s][1] = inb[icoff_n + off[s][1]];
        }
#pragma unroll
        for (int s = 0; s < NSTEP; ++s) {
            v2f a;
            a.x = okm[s][0] ? va[s][0] : 0.0f;
            a.y = okm[s][1] ? va[s][1] : 0.0f;
            const int kb = lbase + (s * 4 + half * 2) * NC + mcol;
#pragma unroll
            for (int t = 0; t < NT; ++t) {
                v2f b;
                b.x = ldsw[kb + t * 16];
                b.y = ldsw[kb + NC + t * 16];
                acc[t] = __builtin_amdgcn_wmma_f32_16x16x4_f32(
                             false, a, false, b, (short)0, acc[t], false, false);
            }
        }
#pragma unroll
        for (int s = 0; s < NSTEP; ++s) {
            va[s][0] = vb[s][0];
            va[s][1] = vb[s][1];
        }
        icoff = icoff_n;
        lbase += KKP * NC;
    }

#pragma unroll
    for (int t = 0; t < NT; ++t) {
#pragma unroll
        for (int r = 0; r < 8; ++r) {
            int prow = mbase + r + half * 8;
            if (prow < npix) {
                float v = acc[t][r];
                if (relu) v = fmaxf(v, 0.0f);
                outb[(long)(ocb + t * 16 + mcol) * npix + prow] = v;
            }
        }
    }
}

// ---------------------------------------------------------------------------
// Cost volume: variance across views. View 0 uses ref features (no warp),
// views 1..2 are bilinear gathers at the projected world point.
// ---------------------------------------------------------------------------
__global__ void cost_volume_kernel(const float* __restrict__ feats, const float* __restrict__ C,
                                   const float* __restrict__ wp, float* __restrict__ cost) {
    int n = blockIdx.x * blockDim.x + threadIdx.x;
    if (n >= NPTS) return;
    int d = n / NPIXF;
    int pix = n - d * NPIXF;
    float X = wp[n], Y = wp[NPTS + n], Z = wp[2 * NPTS + n];

    int   fx0[NVIEW - 1], fy0[NVIEW - 1];
    float fwx[NVIEW - 1], fwy[NVIEW - 1];
    for (int v = 1; v < NVIEW; ++v) {
        const float* B  = C + 20 + v * 21;
        const float* Kv = B + 12;
        float cx = B[0] * X + B[1] * Y + B[2] * Z + B[9];
        float cy = B[3] * X + B[4] * Y + B[5] * Z + B[10];
        float cz = B[6] * X + B[7] * Y + B[8] * Z + B[11];
        float uh = Kv[0] * cx + Kv[1] * cy + Kv[2] * cz;
        float vh = Kv[3] * cx + Kv[4] * cy + Kv[5] * cz;
        float zh = Kv[6] * cx + Kv[7] * cy + Kv[8] * cz;
        float zz = fmaxf(zh, 1e-6f);
        float u  = uh / zz;
        float vv = vh / zz;
        float gx = (u  - 0.5f) / (float)(FWW - 1) * 2.0f - 1.0f;
        float gy = (vv - 0.5f) / (float)(FHH - 1) * 2.0f - 1.0f;
        float ix = ((gx + 1.0f) * (float)FWW - 1.0f) * 0.5f;
        float iy = ((gy + 1.0f) * (float)FHH - 1.0f) * 0.5f;
        float fx = floorf(ix), fy = floorf(iy);
        fx0[v - 1] = (int)fx;  fy0[v - 1] = (int)fy;
        fwx[v - 1] = ix - fx;  fwy[v - 1] = iy - fy;
    }

    for (int c = 0; c < 64; ++c) {
        float f0 = feats[(long)c * NPIXF + pix];   // ref view, depth-invariant
        float s = f0, ss = f0 * f0;
        for (int v = 1; v < NVIEW; ++v) {
            const float* fm = feats + (long)v * 64 * NPIXF + (long)c * NPIXF;
            int X0 = fx0[v - 1], Y0 = fy0[v - 1];
            float WX = fwx[v - 1], WY = fwy[v - 1];
            float val = 0.0f;
#pragma unroll
            for (int ty = 0; ty < 2; ++ty)
#pragma unroll
                for (int tx = 0; tx < 2; ++tx) {
                    int xi = X0 + tx, yi = Y0 + ty;
                    float wgt = (tx ? WX : 1.0f - WX) * (ty ? WY : 1.0f - WY);
                    bool ok = (xi >= 0) && (xi < FWW) && (yi >= 0) && (yi < FHH);
                    int xc = imin(imax(xi, 0), FWW - 1);
                    int yc = imin(imax(yi, 0), FHH - 1);
                    float fv = fm[yc * FWW + xc];
                    val += (ok ? fv : 0.0f) * wgt;
                }
            s += val; ss += val * val;
        }
        float mean = s * (1.0f / 3.0f);
        cost[((long)c * DNUM + d) * NPIXF + pix] = ss * (1.0f / 3.0f) - mean * mean;
    }
}

// ---------------------------------------------------------------------------
// Direct 3x3x3 conv3d (pad 1, stride 1 or 2), one thread per output element.
// ---------------------------------------------------------------------------
__global__ void conv3d_kernel(const float* __restrict__ in, const float* __restrict__ w,
                              float* __restrict__ out, int IC, int OC,
                              int ID, int IH, int IW, int OD, int OH, int OW,
                              int stride, int relu) {
    long idx = (long)blockIdx.x * blockDim.x + threadIdx.x;
    long tot = (long)OC * OD * OH * OW;
    if (idx >= tot) return;
    int oc = (int)(idx / ((long)OD * OH * OW));
    long r = idx - (long)oc * OD * OH * OW;
    int oz = (int)(r / (OH * OW));
    int r2 = (int)(r - (long)oz * OH * OW);
    int oy = r2 / OW, ox = r2 - oy * OW;
    float acc = 0.0f;
    for (int ic = 0; ic < IC; ++ic) {
        const float* wb = w + ((long)oc * IC + ic) * 27;
        const float* ib = in + (long)ic * ID * IH * IW;
        for (int kz = 0; kz < 3; ++kz) {
            int zi = oz * stride + kz - 1;
            if (zi < 0 || zi >= ID) continue;
            for (int ky = 0; ky < 3; ++ky) {
                int yi = oy * stride + ky - 1;
                if (yi < 0 || yi >= IH) continue;
                for (int kx = 0; kx < 3; ++kx) {
                    int xi = ox * stride + kx - 1;
                    if (xi < 0 || xi >= IW) continue;
                    acc = fmaf(ib[((long)zi * IH + yi) * IW + xi], wb[(kz * 3 + ky) * 3 + kx], acc);
                }
            }
        }
    }
    if (relu) acc = fmaxf(acc, 0.0f);
    out[idx] = acc;
}

// ---------------------------------------------------------------------------
// Transposed conv3d (lhs_dilation=2, pad (1,2), k=3) -> output = 2x input.
// w layout: (IC, OC, 3, 3, 3); taps need even parity of (o + k - 1).
// ---------------------------------------------------------------------------
__global__ void deconv3d_kernel(const float* __restrict__ in, const float* __restrict__ w,
                                float* __restrict__ out, int IC, int OC,
                                int ID, int IH, int IW, int relu) {
    int OD = ID * 2, OH = IH * 2, OW = IW * 2;
    long idx = (long)blockIdx.x * blockDim.x + threadIdx.x;
    long tot = (long)OC * OD * OH * OW;
    if (idx >= tot) return;
    int oc = (int)(idx / ((long)OD * OH * OW));
    long r = idx - (long)oc * OD * OH * OW;
    int oz = (int)(r / (OH * OW));
    int r2 = (int)(r - (long)oz * OH * OW);
    int oy = r2 / OW, ox = r2 - oy * OW;
    float acc = 0.0f;
    for (int ic = 0; ic < IC; ++ic) {
        const float* ib = in + (long)ic * ID * IH * IW;
        const float* wb = w + ((long)ic * OC + oc) * 27;
        for (int kz = 0; kz < 3; ++kz) {
            int pz = oz + kz - 1;
            if (pz < 0 || (pz & 1) || (pz >> 1) >= ID) continue;
            int zi = pz >> 1;
            for (int ky = 0; ky < 3; ++ky) {
                int py = oy + ky - 1;
                if (py < 0 || (py & 1) || (py >> 1) >= IH) continue;
                int yi = py >> 1;
                for (int kx = 0; kx < 3; ++kx) {
                    int px = ox + kx - 1;
                    if (px < 0 || (px & 1) || (px >> 1) >= IW) continue;
                    int xi = px >> 1;
                    acc = fmaf(ib[((long)zi * IH + yi) * IW + xi],
                               wb[((2 - kz) * 3 + (2 - ky)) * 3 + (2 - kx)], acc);
                }
            }
        }
    }
    if (relu) acc = fmaxf(acc, 0.0f);
    out[idx] = acc;
}

__global__ void add_inplace_kernel(float* __restrict__ a, const float* __restrict__ b, int n) {
    int i = blockIdx.x * blockDim.x + threadIdx.x;
    if (i < n) a[i] += b[i];
}

// ---------------------------------------------------------------------------
// Softmax over depth, expected depth, 4-tap probability map.
// ---------------------------------------------------------------------------
__global__ void depth_regress_kernel(const float* __restrict__ filt, const float* __restrict__ C,
                                     float* __restrict__ depth_map, float* __restrict__ prob_map) {
    int pix = blockIdx.x * blockDim.x + threadIdx.x;
    if (pix >= NPIXF) return;
    float dstart = C[18], dint = C[19];
    float p[DNUM];
    float mx = -1e30f;
    for (int d = 0; d < DNUM; ++d) {
        float v = -filt[(long)d * NPIXF + pix];
        p[d] = v;
        mx = fmaxf(mx, v);
    }
    float sum = 0.0f;
    for (int d = 0; d < DNUM; ++d) {
        float e = __expf(p[d] - mx);
        p[d] = e;
        sum += e;
    }
    float inv = 1.0f / sum;
    float depth = 0.0f;
    for (int d = 0; d < DNUM; ++d) {
        p[d] *= inv;
        depth += (dstart + (float)d * dint) * p[d];
    }
    depth_map[pix] = depth;
    float di = (depth - dstart) / dint;
    int d0 = (int)floorf(di);
    float pm = 0.0f;
    for (int o = -1; o <= 2; ++o) {
        int dd = imin(imax(d0 + o, 0), DNUM - 1);
        pm += p[dd];
    }
    prob_map[pix] = pm;
}

// ---------------------------------------------------------------------------
// Host-side orchestration
// ---------------------------------------------------------------------------
static inline int cdiv(long a, long b) { return (int)((a + b - 1) / b); }

extern "C" void kernel_launch(void* const* d_in, const int* in_sizes, int n_in,
                              void* d_out, int out_size, void* d_ws, size_t ws_size,
                              hipStream_t stream) {
    (void)in_sizes; (void)n_in; (void)out_size; (void)ws_size;

    const float* img = (const float*)d_in[0];
    const float* cam = (const float*)d_in[1];
    const float* w2d[11];
    for (int i = 0; i < 11; ++i) w2d[i] = (const float*)d_in[2 + i];
    const float* w_c0_1 = (const float*)d_in[13];
    const float* w_c1_0 = (const float*)d_in[14];
    const float* w_c1_1 = (const float*)d_in[15];
    const float* w_c2_0 = (const float*)d_in[16];
    const float* w_c2_1 = (const float*)d_in[17];
    const float* w_c3_0 = (const float*)d_in[18];
    const float* w_c3_1 = (const float*)d_in[19];
    const float* w_c6_2 = (const float*)d_in[20];
    const float* w_d4_0 = (const float*)d_in[21];
    const float* w_d5_0 = (const float*)d_in[22];
    const float* w_d6_0 = (const float*)d_in[23];

    float* ws = (float*)d_ws;
    const long SZ_P    = 3L * 8 * 512 * 640;     // 7,864,320 (largest 2D feature map)
    float* CAM   = ws;                            long off = 128;
    float* P0    = ws + off; off += SZ_P;
    float* P1    = ws + off; off += SZ_P;
    float* FEATS = ws + off; off += 3L * 64 * NPIXF;
    float* COST  = ws + off; off += 64L * DNUM * NPIXF;
    float* C0_1  = ws + off; off += 8L  * 48 * 64 * 80;
    float* C1_0  = ws + off; off += 16L * 24 * 32 * 40;
    float* C1_1  = ws + off; off += 16L * 24 * 32 * 40;
    float* C2_0  = ws + off; off += 32L * 12 * 16 * 20;
    float* C2_1  = ws + off; off += 32L * 12 * 16 * 20;
    float* C3_0  = ws + off; off += 64L * 6 * 8 * 10;
    float* C3_1  = ws + off; off += 64L * 6 * 8 * 10;
    float* C4    = ws + off; off += 32L * 12 * 16 * 20;
    float* C5    = ws + off; off += 16L * 24 * 32 * 40;
    float* C6    = ws + off; off += 8L  * 48 * 64 * 80;
    float* FILT  = ws + off; off += (long)DNUM * NPIXF;

    float* wp        = (float*)d_out;            // world_points: 3*NPTS
    float* depth_map = wp + 3L * NPTS;           // NPIXF
    float* prob_map  = depth_map + NPIXF;        // NPIXF

    // --- camera math + world points ---
    camera_setup_kernel<<<1, 32, 0, stream>>>(cam, CAM);
    world_points_kernel<<<cdiv(NPTS, 256), 256, 0, stream>>>(CAM, wp);

    // --- 2D conv tower (3 images) ---
    conv2d_direct_kernel<<<cdiv(3L * 8 * 512 * 640, 256), 256, 0, stream>>>(
        img, w2d[0], P0, 3, 3, 8, 512, 640, 3, 1);
    conv2d_direct_kernel<<<cdiv(3L * 8 * 512 * 640, 256), 256, 0, stream>>>(
        P0, w2d[1], P1, 3, 8, 8, 512, 640, 3, 1);

    // WMMA layers: block=256 (8 waves), grid=(Mtiles/8, OC/(16*NT), 3).
    // Dynamic LDS = IC*KKP*16*NT floats (zero-padded weight tile).
    {
        // L2: 8->16, 5x5 s2, 512x640 -> 256x320  (NT=1, KKP=28)
        size_t sh = (size_t)8 * 28 * 16 * sizeof(float);
        conv2d_wmma_kernel<5, 1><<<dim3(cdiv(256L * 320, 128), 1, 3), 256, sh, stream>>>(
            P1, w2d[2], P0, 8, 512, 640, 256, 320, 2, 1);
    }
    {
        // L3/L4: 16->16, 3x3 s1 @256x320  (NT=1, KKP=12)
        size_t sh = (size_t)16 * 12 * 16 * sizeof(float);
        conv2d_wmma_kernel<3, 1><<<dim3(cdiv(256L * 320, 128), 1, 3), 256, sh, stream>>>(
            P0, w2d[3], P1, 16, 256, 320, 256, 320, 1, 1);
        conv2d_wmma_kernel<3, 1><<<dim3(cdiv(256L * 320, 128), 1, 3), 256, sh, stream>>>(
            P1, w2d[4], P0, 16, 256, 320, 256, 320, 1, 1);
    }
    {
        // L5: 16->32, 5x5 s2, 256x320 -> 128x160  (NT=2, 32 oc/block)
        size_t sh = (size_t)16 * 28 * 32 * sizeof(float);
        conv2d_wmma_kernel<5, 2><<<dim3(cdiv(128L * 160, 128), 1, 3), 256, sh, stream>>>(
            P0, w2d[5], P1, 16, 256, 320, 128, 160, 2, 1);
    }
    {
        // L6/L7: 32->32, 3x3 s1 @128x160  (NT=2)
        size_t sh = (size_t)32 * 12 * 32 * sizeof(float);
        conv2d_wmma_kernel<3, 2><<<dim3(cdiv(128L * 160, 128), 1, 3), 256, sh, stream>>>(
            P1, w2d[6], P0, 32, 128, 160, 128, 160, 1, 1);
        conv2d_wmma_kernel<3, 2><<<dim3(cdiv(128L * 160, 128), 1, 3), 256, sh, stream>>>(
            P0, w2d[7], P1, 32, 128, 160, 128, 160, 1, 1);
    }
    {
        // L8: 32->64, 5x5 s2, 128x160 -> 64x80  (NT=2, grid.y=2)
        size_t sh = (size_t)32 * 28 * 32 * sizeof(float);
        conv2d_wmma_kernel<5, 2><<<dim3(cdiv(64L * 80, 128), 2, 3), 256, sh, stream>>>(
            P1, w2d[8], P0, 32, 128, 160, 64, 80, 2, 1);
    }
    {
        // L9/L10: 64->64, 3x3 s1 @64x80  (NT=2, grid.y=2)
        size_t sh = (size_t)64 * 12 * 32 * sizeof(float);
        conv2d_wmma_kernel<3, 2><<<dim3(cdiv(64L * 80, 128), 2, 3), 256, sh, stream>>>(
            P0, w2d[9], P1, 64, 64, 80, 64, 80, 1, 1);
        conv2d_wmma_kernel<3, 2><<<dim3(cdiv(64L * 80, 128), 2, 3), 256, sh, stream>>>(
            P1, w2d[10], FEATS, 64, 64, 80, 64, 80, 1, 0);   // no relu
    }

    // --- cost volume (variance across views) ---
    cost_volume_kernel<<<cdiv(NPTS, 256), 256, 0, stream>>>(FEATS, CAM, wp, COST);

    // --- 3D U-Net ---
    conv3d_kernel<<<cdiv(8L * 48 * 64 * 80, 256), 256, 0, stream>>>(
        COST, w_c0_1, C0_1, 64, 8, 48, 64, 80, 48, 64, 80, 1, 1);
    conv3d_kernel<<<cdiv(16L * 24 * 32 * 40, 256), 256, 0, stream>>>(
        COST, w_c1_0, C1_0, 64, 16, 48, 64, 80, 24, 32, 40, 2, 1);
    conv3d_kernel<<<cdiv(32L * 12 * 16 * 20, 256), 256, 0, stream>>>(
        C1_0, w_c2_0, C2_0, 16, 32, 24, 32, 40, 12, 16, 20, 2, 1);
    conv3d_kernel<<<cdiv(64L * 6 * 8 * 10, 256), 256, 0, stream>>>(
        C2_0, w_c3_0, C3_0, 32, 64, 12, 16, 20, 6, 8, 10, 2, 1);
    conv3d_kernel<<<cdiv(16L * 24 * 32 * 40, 256), 256, 0, stream>>>(
        C1_0, w_c1_1, C1_1, 16, 16, 24, 32, 40, 24, 32, 40, 1, 1);
    conv3d_kernel<<<cdiv(32L * 12 * 16 * 20, 256), 256, 0, stream>>>(
        C2_0, w_c2_1, C2_1, 32, 32, 12, 16, 20, 12, 16, 20, 1, 1);
    conv3d_kernel<<<cdiv(64L * 6 * 8 * 10, 256), 256, 0, stream>>>(
        C3_0, w_c3_1, C3_1, 64, 64, 6, 8, 10, 6, 8, 10, 1, 1);

    deconv3d_kernel<<<cdiv(32L * 12 * 16 * 20, 256), 256, 0, stream>>>(
        C3_1, w_d4_0, C4, 64, 32, 6, 8, 10, 1);
    add_inplace_kernel<<<cdiv(32L * 12 * 16 * 20, 256), 256, 0, stream>>>(
        C4, C2_1, 32 * 12 * 16 * 20);
    deconv3d_kernel<<<cdiv(16L * 24 * 32 * 40, 256), 256, 0, stream>>>(
        C4, w_d5_0, C5, 32, 16, 12, 16, 20, 1);
    add_inplace_kernel<<<cdiv(16L * 24 * 32 * 40, 256), 256, 0, stream>>>(
        C5, C1_1, 16 * 24 * 32 * 40);
    deconv3d_kernel<<<cdiv(8L * 48 * 64 * 80, 256), 256, 0, stream>>>(
        C5, w_d6_0, C6, 16, 8, 24, 32, 40, 1);
    add_inplace_kernel<<<cdiv(8L * 48 * 64 * 80, 256), 256, 0, stream>>>(
        C6, C0_1, 8 * 48 * 64 * 80);
    conv3d_kernel<<<cdiv(1L * 48 * 64 * 80, 256), 256, 0, stream>>>(
        C6, w_c6_2, FILT, 8, 1, 48, 64, 80, 48, 64, 80, 1, 0);

    // --- softmax over depth + regression ---
    depth_regress_kernel<<<cdiv(NPIXF, 256), 256, 0, stream>>>(FILT, CAM, depth_map, prob_map);
}